// SPD_Loss_74990128988581
// MI455X (gfx1250) — compile-verified
//
#include <hip/hip_runtime.h>

// SPD loss: 18-bin histogram over 2^24 int32 samples -> scalar f32 loss.
// Memory-bound: 128 MiB @ 23.3 TB/s ~= 5.5 us. Design:
//   K1: async-to-LDS staged streaming + lane-private LDS histogram (ds_add_u32,
//       conflict-free layout), per-block f32 partials to ws.
//   K2: column-sum of partials via v_wmma_f32_16x16x4_f32 (A = ones), then loss.

#define N_CLASSES 9
#define N_GROUPS  2
#define NBINS     18          // 9*2
#define BLOCK     256
#define WAVES     8
#define LANES     32
#define STAGE     1024        // elements per pipeline stage = BLOCK * 4
#define GRID      1024

typedef __attribute__((ext_vector_type(2))) float v2f;
typedef __attribute__((ext_vector_type(8))) float v8f;

// ---- CDNA5 async-to-LDS helpers (ASYNCcnt-tracked). Guarded so the host
// pass of hipcc never sees gcn asm. ----
__device__ __forceinline__ void async_copy16(unsigned lds_off, const int* gaddr) {
#if defined(__HIP_DEVICE_COMPILE__)
  // VDST = per-lane LDS byte offset, VADDR = 64-bit global address (GV mode)
  asm volatile("global_load_async_to_lds_b128 %0, %1, off"
               :: "v"(lds_off), "v"(gaddr) : "memory");
#else
  (void)lds_off; (void)gaddr;
#endif
}
__device__ __forceinline__ void wait_async0() {
#if defined(__HIP_DEVICE_COMPILE__)
  asm volatile("s_wait_asynccnt 0" ::: "memory");
#endif
}
__device__ __forceinline__ void wait_async2() {
#if defined(__HIP_DEVICE_COMPILE__)
  asm volatile("s_wait_asynccnt 2" ::: "memory");
#endif
}

// ---------------- Kernel 1: streaming histogram ----------------
__global__ __launch_bounds__(BLOCK) void spd_hist_kernel(
    const int* __restrict__ preds, const int* __restrict__ attrs,
    float* __restrict__ partial, long long n)
{
  __shared__ __align__(16) int sPred[2][STAGE];   // 8 KB
  __shared__ __align__(16) int sAttr[2][STAGE];   // 8 KB
  __shared__ unsigned hist[WAVES * NBINS * LANES]; // 18 KB lane-private bins
  __shared__ unsigned bhist[NBINS];

  const int tid  = threadIdx.x;
  const int wave = tid >> 5;
  const int lane = tid & 31;

  for (int i = tid; i < WAVES * NBINS * LANES; i += BLOCK) hist[i] = 0u;
  if (tid < NBINS) bhist[tid] = 0u;
  __syncthreads();

  // bank(addr) = (32*bin + lane) mod 64 -> conflict-free within a wave.
  unsigned* myhist = &hist[wave * NBINS * LANES + lane];

  const long long nStages = n / STAGE;
  long long s = blockIdx.x;
  int buf = 0;

  if (s < nStages) {
    async_copy16((unsigned)(unsigned long long)&sPred[0][tid * 4],
                 preds + s * STAGE + tid * 4);
    async_copy16((unsigned)(unsigned long long)&sAttr[0][tid * 4],
                 attrs + s * STAGE + tid * 4);
  }

  while (s < nStages) {
    const long long nx = s + (long long)gridDim.x;
    if (nx < nStages) {
      async_copy16((unsigned)(unsigned long long)&sPred[buf ^ 1][tid * 4],
                   preds + nx * STAGE + tid * 4);
      async_copy16((unsigned)(unsigned long long)&sAttr[buf ^ 1][tid * 4],
                   attrs + nx * STAGE + tid * 4);
      wait_async2();   // current stage's 2 async loads are complete
    } else {
      wait_async0();
    }
    // Each thread consumes exactly the 4 elements it staged: no barrier needed.
    const int4 p = *(const int4*)&sPred[buf][tid * 4];
    const int4 a = *(const int4*)&sAttr[buf][tid * 4];
    atomicAdd(myhist + (unsigned)(p.x * N_GROUPS + a.x) * LANES, 1u);
    atomicAdd(myhist + (unsigned)(p.y * N_GROUPS + a.y) * LANES, 1u);
    atomicAdd(myhist + (unsigned)(p.z * N_GROUPS + a.z) * LANES, 1u);
    atomicAdd(myhist + (unsigned)(p.w * N_GROUPS + a.w) * LANES, 1u);
    buf ^= 1;
    s = nx;
  }

  // Tail (n % STAGE) via direct loads; empty for N = 2^24.
  for (long long i = nStages * STAGE + (long long)blockIdx.x * BLOCK + tid;
       i < n; i += (long long)gridDim.x * BLOCK) {
    const int b = preds[i] * N_GROUPS + attrs[i];
    atomicAdd(myhist + (unsigned)b * LANES, 1u);
  }
  __syncthreads();

  // Reduce 256 lane-private histograms -> 18 block counts.
  for (int b = 0; b < NBINS; ++b) {
    unsigned v = hist[wave * NBINS * LANES + b * LANES + lane];
#pragma unroll
    for (int off = 16; off > 0; off >>= 1) v += __shfl_down(v, off, 32);
    if (lane == 0) atomicAdd(&bhist[b], v);
  }
  __syncthreads();

  // Partial row: 32 floats, cols 18..31 zero (clean WMMA operand padding).
  if (tid < 32) {
    const float val = (tid < NBINS) ? (float)bhist[tid] : 0.0f;
    partial[(long long)blockIdx.x * 32 + tid] = val;
  }
}

// ---------------- Kernel 2: WMMA column-sum + loss ----------------
__global__ __launch_bounds__(BLOCK) void spd_loss_kernel(
    const float* __restrict__ partial, float* __restrict__ out,
    int nrows, float Nf)
{
  __shared__ float sums[32];
  const int tid  = threadIdx.x;
  const int wave = tid >> 5;
  const int lane = tid & 31;
  if (tid < 32) sums[tid] = 0.0f;
  __syncthreads();

  const int rowsPerWave = nrows / WAVES;   // nrows is a multiple of 32
  const int rbeg = wave * rowsPerWave;
  const int rend = rbeg + rowsPerWave;

#if __has_builtin(__builtin_amdgcn_wmma_f32_16x16x4_f32)
  // D = A x B + C with A = ones(16x4)  =>  D[m][n] = sum_k B[k][n].
  // B 4x16 f32 layout: VGPR0 = {row0 | row2}, VGPR1 = {row1 | row3},
  // lanes 0-15 -> N = lane, lanes 16-31 -> N = lane-16.
  v8f c0 = {0.f, 0.f, 0.f, 0.f, 0.f, 0.f, 0.f, 0.f};
  v8f c1 = c0;
  const v2f ones = {1.0f, 1.0f};
  const int half = lane >> 4;
  const int col  = lane & 15;
  for (int g = rbeg; g < rend; g += 4) {
    const int r0 = g + half * 2;
    v2f b0, b1;
    b0.x = partial[r0 * 32 + col];
    b0.y = partial[(r0 + 1) * 32 + col];
    b1.x = partial[r0 * 32 + 16 + col];     // bins 16..17 (+ zero padding)
    b1.y = partial[(r0 + 1) * 32 + 16 + col];
    c0 = __builtin_amdgcn_wmma_f32_16x16x4_f32(false, ones, false, b0,
                                               (short)0, c0, false, false);
    c1 = __builtin_amdgcn_wmma_f32_16x16x4_f32(false, ones, false, b1,
                                               (short)0, c1, false, false);
  }
  // D row 0: VGPR0, lanes 0-15 hold column sums (all rows identical).
  if (lane < 16) atomicAdd(&sums[lane], c0[0]);
  if (lane < 2)  atomicAdd(&sums[16 + lane], c1[0]);
#else
  float acc = 0.0f;
  for (int r = rbeg; r < rend; ++r) acc += partial[r * 32 + lane];
  atomicAdd(&sums[lane], acc);
#endif
  __syncthreads();

  if (tid == 0) {
    // sums[2c]=count(class c, group 0), sums[2c+1]=count(class c, group 1).
    float n1 = 0.0f;
    for (int c = 0; c < N_CLASSES; ++c) n1 += sums[2 * c + 1];
    const float n0 = Nf - n1;
    float loss = 0.0f;
    for (int c = 0; c < N_CLASSES; ++c) {
      const float d = sums[2 * c] / n0 - sums[2 * c + 1] / n1;
      loss += d * d;
    }
    out[0] = loss;
  }
}

extern "C" void kernel_launch(void* const* d_in, const int* in_sizes, int n_in,
                              void* d_out, int out_size, void* d_ws, size_t ws_size,
                              hipStream_t stream) {
  (void)n_in; (void)out_size;
  const int* preds = (const int*)d_in[0];
  const int* attrs = (const int*)d_in[1];
  float* out      = (float*)d_out;
  float* partial  = (float*)d_ws;          // GRID * 32 floats (128 KB)
  const long long n = (long long)in_sizes[0];

  int grid = GRID;
  if ((size_t)grid * 32 * sizeof(float) > ws_size) {
    grid = (int)(ws_size / (32 * sizeof(float)));
    grid &= ~31;                            // keep multiple of 32
    if (grid < 32) grid = 32;
  }
  spd_hist_kernel<<<grid, BLOCK, 0, stream>>>(preds, attrs, partial, n);
  spd_loss_kernel<<<1, BLOCK, 0, stream>>>(partial, out, grid, (float)n);
}